// CapsNet_76244259439241
// MI455X (gfx1250) — compile-verified
//
#include <hip/hip_runtime.h>
#include <hip/hip_bf16.h>
#include <math.h>

// CapsNet forward for MI455X (gfx1250, wave32).
// All GEMM-shaped work (3 convs as implicit GEMM + 3 decoder FCs) uses
// V_WMMA_F32_16X16X4_F32 (full-precision f32 matrix op). Each wave owns a
// 32(M) x 32(N) C tile: 4 WMMAs per K-step. B-fragment addresses are kept as
// live pointers bumped by an incremental im2col delta (two carry-selects per
// k-stream) -- no divides, no exec branching, no 64-bit address rebuilds in
// the hot loop.
// Dynamic routing is fused: b_log == u_hat * cumsum(v), so neither b_log nor
// u_hat (158 MB each) is ever materialized; u_hat is recomputed per pass.

typedef __attribute__((ext_vector_type(2))) float v2f;
typedef __attribute__((ext_vector_type(8))) float v8f;

// Incremental im2col k-stream: tracks (kx, ky) for one k index advancing by
// +4 per step and yields the element delta. Requires KW >= 4 (true: 5, 9, 9).
struct KStream {
    int kx, ky;
};

__device__ __forceinline__ int ks_init(KStream& s, int k, int khw, int KW,
                                       int IH, int IW) {
    const int ic = k / khw;
    const int r  = k - ic * khw;
    s.ky = r / KW;
    s.kx = r - s.ky * KW;
    return ic * IH * IW + s.ky * IW + s.kx;   // initial element offset
}

__device__ __forceinline__ int ks_delta4(KStream& s, int KW, int KH,
                                         int rowAdj, int chAdj) {
    // rowAdj = IW - KW ; chAdj = (IH - KH) * IW ; returns element delta for +4.
    s.kx += 4;
    int d = 4;
    const bool rc = s.kx >= KW;
    s.kx = rc ? s.kx - KW : s.kx;
    d    = rc ? d + rowAdj : d;
    s.ky = rc ? s.ky + 1 : s.ky;
    const bool cc = rc & (s.ky == KH);
    s.ky = cc ? 0 : s.ky;
    d    = cc ? d + chAdj : d;
    return d;
}

#define WMMA_F32(A, Bf, C) \
    __builtin_amdgcn_wmma_f32_16x16x4_f32(false, (A), false, (Bf), (short)0, (C), false, false)

// ---------------------------------------------------------------------------
// Implicit-GEMM convolution, fp32 WMMA.
//   C[oc, p] = sum_k W[oc, k] * patch[k, p],  k = (ic,ky,kx), p = (b,oy,ox)
// grid = (numPixels/32, OC/32), block = 32 (one wave -> one 32x32 C tile).
// ---------------------------------------------------------------------------
__global__ __launch_bounds__(32)
void wmma_conv_kernel(const float* __restrict__ in, const float* __restrict__ w,
                      const float* __restrict__ bias, float* __restrict__ out,
                      int IC, int IH, int IW, int OC, int KH, int KW,
                      int OH, int OW, int stride, int relu, int K)
{
    const int lane = threadIdx.x;          // 0..31
    const int m0   = blockIdx.y << 5;      // oc tile base (2 x 16)
    const int n0   = blockIdx.x << 5;      // pixel tile base (2 x 16)
    const int l16  = lane & 15;
    const int koff = (lane >> 4) << 1;     // 0 (lanes 0-15) or 2 (lanes 16-31)

    // Decode this lane's two output pixels (columns of the two B fragments).
    const int ohw = OH * OW;
    int bimg[2], yx[2];
    const float* __restrict__ inb[2];
#pragma unroll
    for (int t = 0; t < 2; ++t) {
        const int p = n0 + (t << 4) + l16;
        bimg[t] = p / ohw;
        yx[t]   = p - bimg[t] * ohw;
        const int oy = yx[t] / OW;
        const int ox = yx[t] - oy * OW;
        inb[t] = in + ((size_t)bimg[t] * IC * IH + (size_t)oy * stride) * IW
                    + (size_t)ox * stride;
    }
    // A fragment rows: oc = m0 + l16 and m0 + 16 + l16; weights row-major [OC][K].
    const float* __restrict__ wrow0 = w + (size_t)(m0 + l16) * K;
    const float* __restrict__ wrow1 = w + (size_t)(m0 + 16 + l16) * K;

    const int khw    = KH * KW;
    const int rowAdj = IW - KW;
    const int chAdj  = (IH - KH) * IW;

    // Two k-streams per lane (k = k0+koff and k0+koff+1), shared by all tiles.
    KStream s0, s1;
    const int off0 = ks_init(s0, koff,     khw, KW, IH, IW);
    const int off1 = ks_init(s1, koff + 1, khw, KW, IH, IW);

    // Live B pointers: pB[tile][stream].
    const float* pB00 = inb[0] + off0;
    const float* pB01 = inb[0] + off1;
    const float* pB10 = inb[1] + off0;
    const float* pB11 = inb[1] + off1;

    v8f c00 = {0.f,0.f,0.f,0.f,0.f,0.f,0.f,0.f};   // (m-tile 0, n-tile 0)
    v8f c01 = {0.f,0.f,0.f,0.f,0.f,0.f,0.f,0.f};   // (m-tile 0, n-tile 1)
    v8f c10 = {0.f,0.f,0.f,0.f,0.f,0.f,0.f,0.f};   // (m-tile 1, n-tile 0)
    v8f c11 = {0.f,0.f,0.f,0.f,0.f,0.f,0.f,0.f};   // (m-tile 1, n-tile 1)

    // Main loop: no guards (all k in range). Lanes touch up to k0+3.
    int k0 = 0;
    for (; k0 + 4 <= K; k0 += 4) {
        // A 16x4 f32: lanes 0-15 hold K=k0,k0+1; lanes 16-31 hold k0+2,k0+3.
        const v2f a0 = *(const v2f*)(wrow0 + k0 + koff);   // 8B aligned only when K is even (conv1 K=81: 4B)
        const v2f a1 = *(const v2f*)(wrow1 + k0 + koff);
        v2f b0, b1;
        b0.x = *pB00;  b0.y = *pB01;
        b1.x = *pB10;  b1.y = *pB11;
        c00 = WMMA_F32(a0, b0, c00);
        c01 = WMMA_F32(a0, b1, c01);
        c10 = WMMA_F32(a1, b0, c10);
        c11 = WMMA_F32(a1, b1, c11);
        const int d0 = ks_delta4(s0, KW, KH, rowAdj, chAdj);
        const int d1 = ks_delta4(s1, KW, KH, rowAdj, chAdj);
        pB00 += d0;  pB10 += d0;
        pB01 += d1;  pB11 += d1;
    }

    // Tail (only if K % 4 != 0, e.g. conv1 K=81): one guarded WMMA quad.
    if (k0 < K) {
        const bool g0 = (k0 + koff)     < K;
        const bool g1 = (k0 + koff + 1) < K;
        v2f a0, a1, b0, b1;
        a0.x = g0 ? wrow0[k0 + koff]     : 0.f;
        a0.y = g1 ? wrow0[k0 + koff + 1] : 0.f;
        a1.x = g0 ? wrow1[k0 + koff]     : 0.f;
        a1.y = g1 ? wrow1[k0 + koff + 1] : 0.f;
        b0.x = g0 ? *pB00 : 0.f;
        b0.y = g1 ? *pB01 : 0.f;
        b1.x = g0 ? *pB10 : 0.f;
        b1.y = g1 ? *pB11 : 0.f;
        c00 = WMMA_F32(a0, b0, c00);
        c01 = WMMA_F32(a0, b1, c01);
        c10 = WMMA_F32(a1, b0, c10);
        c11 = WMMA_F32(a1, b1, c11);
    }

    // C/D layout: vgpr v -> M = v + 8*(lane>=16), N = lane&15.
    const int mhalf = (lane >> 4) << 3;
#pragma unroll
    for (int mt = 0; mt < 2; ++mt) {
        const int mb = m0 + (mt << 4) + mhalf;
#pragma unroll
        for (int t = 0; t < 2; ++t) {
            float* __restrict__ outp = out + (size_t)bimg[t] * OC * ohw + yx[t];
            const v8f cv = mt ? (t ? c11 : c10) : (t ? c01 : c00);
#pragma unroll
            for (int vv = 0; vv < 8; ++vv) {
                const int m = mb + vv;
                float val = cv[vv] + bias[m];
                if (relu) val = fmaxf(val, 0.f);
                outp[(size_t)m * ohw] = val;
            }
        }
    }
}

// ---------------------------------------------------------------------------
// Decoder FC: out[16 x N] = act(A[16 x K] @ W[K x N] + bias[N])
// grid = N/16, block = 32. act: 0 = relu, 1 = sigmoid. Requires K % 4 == 0.
// ---------------------------------------------------------------------------
__global__ __launch_bounds__(32)
void wmma_fc_kernel(const float* __restrict__ A, const float* __restrict__ W,
                    const float* __restrict__ bias, float* __restrict__ out,
                    int K, int N, int act)
{
    const int lane = threadIdx.x;
    const int n0   = blockIdx.x << 4;
    const int l16  = lane & 15;
    const int koff = (lane >> 4) << 1;
    const int n    = n0 + l16;

    const float* __restrict__ arow = A + (size_t)l16 * K + koff;
    const float* __restrict__ wcol = W + (size_t)koff * N + n;
    const size_t wstep = (size_t)4 * N;

    v8f c = {0.f,0.f,0.f,0.f,0.f,0.f,0.f,0.f};
    for (int k0 = 0; k0 < K; k0 += 4) {
        const v2f a = *(const v2f*)(arow + k0);          // 8B aligned: K even
        v2f b;
        b.x = wcol[0];
        b.y = wcol[N];
        c = WMMA_F32(a, b, c);
        wcol += wstep;
    }

    const int mb = (lane >> 4) << 3;
#pragma unroll
    for (int vv = 0; vv < 8; ++vv) {
        const int m = mb + vv;
        float val = c[vv] + bias[n];
        val = act ? (1.f / (1.f + expf(-val))) : fmaxf(val, 0.f);
        out[(size_t)m * N + n] = val;
    }
}

// ---------------------------------------------------------------------------
// squash over last dim (8), in place. One thread per capsule (b,i).
// ---------------------------------------------------------------------------
__global__ void squash8_kernel(float* __restrict__ u, int ncaps)
{
    const int idx = blockIdx.x * blockDim.x + threadIdx.x;
    if (idx >= ncaps) return;
    float* __restrict__ p = u + (size_t)idx * 8;
    float x[8]; float s2 = 0.f;
#pragma unroll
    for (int k = 0; k < 8; ++k) { x[k] = p[k]; s2 += x[k] * x[k]; }
    const float sc = s2 / (1.f + s2) / sqrtf(s2 + 1e-8f);
#pragma unroll
    for (int k = 0; k < 8; ++k) p[k] = x[k] * sc;
}

__global__ void zero_kernel(float* __restrict__ dst, int n)
{
    const int idx = blockIdx.x * blockDim.x + threadIdx.x;
    if (idx < n) dst[idx] = 0.f;
}

// ---------------------------------------------------------------------------
// Routing pass (fused): recomputes u_hat on the fly.
//   logits[b,i,j,d] = u_hat * Vacc[b,j,d]   (Vacc = sum of previous v's)
//   c = softmax_j(logits);  partial_s[b,j,d] += c * u_hat over an i-chunk.
// block = 256 = (d:16) x (b:16); grid.x = NI/RT_IT chunks. Deterministic:
// partials written per chunk, summed by routing_reduce_kernel.
// ---------------------------------------------------------------------------
#define RT_IT 32
__global__ __launch_bounds__(256)
void routing_pass_kernel(const float* __restrict__ u,     // [16][NI][8]
                         const float* __restrict__ Wc,    // [10][NI][16][8]
                         const float* __restrict__ Vacc,  // [16][10][16]
                         float* __restrict__ partial,     // [chunks][16][10][16]
                         int NI)
{
    const int d  = threadIdx.x & 15;
    const int b  = threadIdx.x >> 4;
    const int i0 = blockIdx.x * RT_IT;

    float vj[10];
#pragma unroll
    for (int j = 0; j < 10; ++j) vj[j] = Vacc[(b * 10 + j) * 16 + d];
    float acc[10];
#pragma unroll
    for (int j = 0; j < 10; ++j) acc[j] = 0.f;

    for (int it = 0; it < RT_IT; ++it) {
        const int i = i0 + it;
        float uk[8];
        const float* __restrict__ up = u + ((size_t)b * NI + i) * 8;
#pragma unroll
        for (int k = 0; k < 8; ++k) uk[k] = up[k];

        float uh[10], lg[10];
        float mx = -3.4e38f;
#pragma unroll
        for (int j = 0; j < 10; ++j) {
            const float* __restrict__ wp =
                Wc + ((((size_t)j * NI + i) * 16 + d) << 3);
            float s = 0.f;
#pragma unroll
            for (int k = 0; k < 8; ++k) s += wp[k] * uk[k];
            uh[j] = s;
            const float l = s * vj[j];
            lg[j] = l;
            mx = fmaxf(mx, l);
        }
        float se = 0.f, e[10];
#pragma unroll
        for (int j = 0; j < 10; ++j) { e[j] = expf(lg[j] - mx); se += e[j]; }
        const float inv = 1.f / se;
#pragma unroll
        for (int j = 0; j < 10; ++j) acc[j] += (e[j] * inv) * uh[j];
    }

#pragma unroll
    for (int j = 0; j < 10; ++j)
        partial[(((size_t)blockIdx.x * 16 + b) * 10 + j) * 16 + d] = acc[j];
}

// Sum partials -> s[b,j,d]; squash over d (16); write v; optionally Vacc += v.
// grid = (10, 16) = (j, b); block = 16 (d).
__global__ void routing_reduce_kernel(const float* __restrict__ partial,
                                      float* __restrict__ v,
                                      float* __restrict__ Vacc,
                                      int nchunks, int updateV)
{
    const int j = blockIdx.x, b = blockIdx.y, d = threadIdx.x;
    float s = 0.f;
    for (int c = 0; c < nchunks; ++c)
        s += partial[(((size_t)c * 16 + b) * 10 + j) * 16 + d];
    __shared__ float sd[16];
    sd[d] = s * s;
    __syncthreads();
    float s2 = 0.f;
#pragma unroll
    for (int k = 0; k < 16; ++k) s2 += sd[k];
    const float sc = s2 / (1.f + s2) / sqrtf(s2 + 1e-8f);
    const float vd = s * sc;
    v[(b * 10 + j) * 16 + d] = vd;
    if (updateV) Vacc[(b * 10 + j) * 16 + d] += vd;
}

// out_caps = ||v||_d ; argmax mask ; masked (16 x 160). grid = 16, block = 32.
__global__ void caps_out_kernel(const float* __restrict__ v,
                                float* __restrict__ out_caps,
                                float* __restrict__ masked)
{
    const int b = blockIdx.x, t = threadIdx.x;
    __shared__ float norms[10];
    __shared__ int best;
    if (t < 10) {
        float s2 = 0.f;
        for (int d = 0; d < 16; ++d) {
            const float x = v[(b * 10 + t) * 16 + d];
            s2 += x * x;
        }
        norms[t] = sqrtf(s2);
        out_caps[b * 10 + t] = norms[t];
    }
    __syncthreads();
    if (t == 0) {
        int bi = 0; float bv = norms[0];
        for (int j = 1; j < 10; ++j)
            if (norms[j] > bv) { bv = norms[j]; bi = j; }   // first-max, like argmax
        best = bi;
    }
    __syncthreads();
    for (int idx = t; idx < 160; idx += 32) {
        const int j = idx >> 4;
        masked[b * 160 + idx] = (j == best) ? v[b * 160 + idx] : 0.f;
    }
}

// ---------------------------------------------------------------------------
extern "C" void kernel_launch(void* const* d_in, const int* in_sizes, int n_in,
                              void* d_out, int out_size, void* d_ws, size_t ws_size,
                              hipStream_t stream)
{
    (void)in_sizes; (void)n_in; (void)out_size; (void)ws_size;
    const float* x   = (const float*)d_in[0];
    const float* c1w = (const float*)d_in[1];
    const float* c1b = (const float*)d_in[2];
    const float* c2w = (const float*)d_in[3];
    const float* c2b = (const float*)d_in[4];
    const float* pcw = (const float*)d_in[5];
    const float* pcb = (const float*)d_in[6];
    const float* Wc  = (const float*)d_in[7];
    const float* w1  = (const float*)d_in[8];
    const float* b1  = (const float*)d_in[9];
    const float* w2  = (const float*)d_in[10];
    const float* b2  = (const float*)d_in[11];
    const float* w3  = (const float*)d_in[12];
    const float* b3  = (const float*)d_in[13];
    float* out = (float*)d_out;
    float* ws  = (float*)d_ws;

    const int B = 16, NI = 15488;

    // Workspace layout (floats). h1/h2 die before routing -> partials overlay.
    const size_t P_OFF = 0;                         // p / u : 16*256*22*22 = 1,982,464
    const size_t BIG   = 1982464;
    const size_t H1    = BIG;                       // 16*256*56*56 = 12,845,056
    const size_t H2    = BIG + 12845056;            // 16*256*52*52 = 11,075,584
    const size_t PART  = BIG;                       // 484*16*10*16 = 1,239,040
    const size_t VACC  = PART + 1239040;            // 2560
    const size_t VBUF  = VACC + 2560;               // 2560
    const size_t MASKD = VBUF + 2560;               // 2560
    const size_t D1    = MASKD + 2560;              // 16*512  = 8192
    const size_t D2    = D1 + 8192;                 // 16*1024 = 16384

    float* p      = ws + P_OFF;
    float* h1     = ws + H1;
    float* h2     = ws + H2;
    float* part   = ws + PART;
    float* Vacc   = ws + VACC;
    float* vbuf   = ws + VBUF;
    float* masked = ws + MASKD;
    float* d1     = ws + D1;
    float* d2     = ws + D2;

    // conv1: 1->256, 9x9 s1, 64->56, relu. N = 16*56*56 = 50176 -> 1568 x 8 tiles.
    wmma_conv_kernel<<<dim3(1568, 8), 32, 0, stream>>>(
        x, c1w, c1b, h1, /*IC*/1, /*IH*/64, /*IW*/64, /*OC*/256,
        /*KH*/9, /*KW*/9, /*OH*/56, /*OW*/56, /*stride*/1, /*relu*/1, /*K*/81);

    // conv2: 256->256, 5x5 s1, 56->52, relu. N = 16*52*52 = 43264 -> 1352 x 8.
    wmma_conv_kernel<<<dim3(1352, 8), 32, 0, stream>>>(
        h1, c2w, c2b, h2, 256, 56, 56, 256, 5, 5, 52, 52, 1, 1, 6400);

    // primary caps: 256->256, 9x9 s2, 52->22, no relu. N = 16*22*22 = 7744 -> 242 x 8.
    wmma_conv_kernel<<<dim3(242, 8), 32, 0, stream>>>(
        h2, pcw, pcb, p, 256, 52, 52, 256, 9, 9, 22, 22, 2, 0, 20736);

    // squash capsules in place: u = squash(p), 16*15488 capsules of dim 8.
    {
        const int ncaps = B * NI;
        squash8_kernel<<<(ncaps + 255) / 256, 256, 0, stream>>>(p, ncaps);
    }

    // routing: Vacc = 0; 3 fused passes (b_log == u_hat * Vacc).
    zero_kernel<<<(2560 + 255) / 256, 256, 0, stream>>>(Vacc, 2560);
    const int nchunks = NI / RT_IT;   // 484
    for (int r = 0; r < 3; ++r) {
        routing_pass_kernel<<<nchunks, 256, 0, stream>>>(p, Wc, Vacc, part, NI);
        routing_reduce_kernel<<<dim3(10, 16), 16, 0, stream>>>(
            part, vbuf, Vacc, nchunks, (r < 2) ? 1 : 0);
    }

    // out_caps (-> d_out[0:160]) + masked capsule vector.
    caps_out_kernel<<<16, 32, 0, stream>>>(vbuf, out, masked);

    // decoder: 160 -> 512 relu -> 1024 relu -> 4096 sigmoid (-> d_out[160:]).
    wmma_fc_kernel<<<512 / 16, 32, 0, stream>>>(masked, w1, b1, d1, 160, 512, 0);
    wmma_fc_kernel<<<1024 / 16, 32, 0, stream>>>(d1, w2, b2, d2, 512, 1024, 0);
    wmma_fc_kernel<<<4096 / 16, 32, 0, stream>>>(d2, w3, b3, out + 160, 1024, 4096, 1);
}